// FocalLoss_65008624992889
// MI455X (gfx1250) — compile-verified
//
#include <hip/hip_runtime.h>

// Focal loss (RetinaNet classification term) for MI455X / gfx1250.
// Memory-bound: 168 MB of classifications streamed once (~7.2 us at 23.3 TB/s).
// Branchless focal math using raw v_log_f32 (log2, ln2 folded into weights),
// NT B128 streaming loads, LDS-held anchor states, and a
// V_WMMA_F32_16X16X4_F32 ones-B reduction for the wave-level sum.

typedef __attribute__((ext_vector_type(2))) float v2f;
typedef __attribute__((ext_vector_type(4))) float v4f;
typedef __attribute__((ext_vector_type(8))) float v8f;

static constexpr int B_COUNT = 8;
static constexpr int A_COUNT = 65536;
static constexpr int C_COUNT = 80;
static constexpr int N_BOX   = 64;
static constexpr int APB     = 256;   // anchors per block
static constexpr float F_ALPHA = 0.25f;
static constexpr float F_EPS   = 1.0e-4f;
static constexpr float LN2     = 0.69314718055994530942f;

__global__ void focal_init_ws(float* __restrict__ sums, int* __restrict__ counts) {
    int t = threadIdx.x;
    if (t < B_COUNT) { sums[t] = 0.0f; counts[t] = 0; }
}

__global__ __launch_bounds__(256) void focal_main(
    const float* __restrict__ cls,      // (B, A, C)
    const float* __restrict__ anchors,  // (1, A, 2)
    const float* __restrict__ ann,      // (B, N, 3)
    float* __restrict__ sums,
    int*   __restrict__ counts)
{
    __shared__ float sbox[N_BOX * 3];
    __shared__ int   sstate[APB];       // >=0: pos w/ label; -1: neg; -2: ignore

    const int tid = threadIdx.x;
    const int b   = blockIdx.y;
    const int a0  = blockIdx.x * APB;

    // stage this batch's 64 GT boxes (192 floats) into LDS
    if (tid < N_BOX * 3) sbox[tid] = ann[b * N_BOX * 3 + tid];
    __syncthreads();

    // ---- phase 1: per-anchor IoU state (one anchor per thread) ----
    const int a = a0 + tid;
    const float as = anchors[2 * a + 0];
    const float ae = anchors[2 * a + 1];
    const float alen = ae - as;

    float best = -1.0f;
    int   barg = 0;
    #pragma unroll 8
    for (int n = 0; n < N_BOX; ++n) {
        const float bs = sbox[n * 3 + 0];
        const float be = sbox[n * 3 + 1];
        float iw = fminf(ae, be) - fmaxf(as, bs);
        iw = fmaxf(iw, 0.0f);
        const float ua  = fmaxf(alen + (be - bs) - iw, 1.0e-8f);
        const float iou = iw / ua;
        if (iou > best) { best = iou; barg = n; }   // strict >: first-max (argmax semantics)
    }
    const bool pos = best >= 0.5f;
    int st;
    if (pos)                 st = (int)sbox[barg * 3 + 2];
    else if (best < 0.4f)    st = -1;   // negative: all-classes target 0
    else                     st = -2;   // ignore: zero loss
    sstate[tid] = st;

    // num_pos: one atomic per wave32
    unsigned long long bm = __ballot(pos ? 1 : 0);
    if ((tid & 31) == 0) atomicAdd(&counts[b], (int)__popcll(bm));
    __syncthreads();

    // ---- phase 2: stream classifications, coalesced NT float4 loads ----
    // Both focal branches share the form  w * (1-y)^2 * (-ln y):
    //   target 1 (pos, assigned class): y = x,     w = ALPHA
    //   target 0 (pos other / neg):     y = 1 - x, w = 1 - ALPHA
    //   ignore:                         w = 0
    // -ln y = -LN2 * log2 y, and v_log_f32 computes log2 natively, so fold
    // LN2 into the weights and use the raw builtin (x is clamped to
    // [1e-4, 1-1e-4]: no denorm/inf fixups needed).
    const float* base = cls + ((size_t)b * A_COUNT + a0) * C_COUNT;
    float acc = 0.0f;

    // 256 anchors * 80 classes = 20480 floats = 5120 float4 groups / 256 thr = 20 iters
    #pragma unroll 10
    for (int it = 0; it < 20; ++it) {
        const int g    = tid + it * 256;     // float4 group index
        const int flat = g * 4;
        const int al   = flat / C_COUNT;     // local anchor (80 % 4 == 0: no row crossing)
        const int c    = flat - al * C_COUNT;
        const int st2  = sstate[al];

        const v4f v = __builtin_nontemporal_load(
            reinterpret_cast<const v4f*>(base + flat));

        const float wBase = (st2 == -2) ? 0.0f : (1.0f - F_ALPHA) * LN2;
        #pragma unroll
        for (int j = 0; j < 4; ++j) {
            const float x    = fminf(fmaxf(v[j], F_EPS), 1.0f - F_EPS);
            const bool  isP  = (st2 == c + j);              // pos anchor, assigned class
            const float y    = isP ? x : (1.0f - x);        // v_cndmask
            const float w    = isP ? F_ALPHA * LN2 : wBase; // v_cndmask
            const float omy  = 1.0f - y;
            const float lg2  = __builtin_amdgcn_logf(y);    // raw v_log_f32 (log2)
            acc -= (w * omy * omy) * lg2;                   // w' * (1-y)^2 * (-log2 y) * ln2
        }
    }

    // ---- wave32 reduction via V_WMMA_F32_16X16X4_F32 (ones-B trick) ----
    // A (16x4 f32): lane L<16 holds A[L][0]=acc, lanes 16..31 hold A[L-16][2]=acc.
    // B = ones -> D[m][n] = acc_m + acc_{m+16}; lane-local sum of the 8 D VGPRs
    // gives half-wave totals, xor-16 shuffle completes the 32-lane sum.
    v2f amat; amat[0] = acc;  amat[1] = 0.0f;
    v2f bmat; bmat[0] = 1.0f; bmat[1] = 1.0f;
    v8f cmat = {};
    v8f d = __builtin_amdgcn_wmma_f32_16x16x4_f32(
        /*neg_a=*/false, amat, /*neg_b=*/false, bmat,
        /*c_mod=*/(short)0, cmat, /*reuse_a=*/false, /*reuse_b=*/false);

    float t = d[0] + d[1] + d[2] + d[3] + d[4] + d[5] + d[6] + d[7];
    const float wsum = t + __shfl_xor(t, 16, 32);
    if ((tid & 31) == 0) atomicAdd(&sums[b], wsum);
}

__global__ void focal_finalize(const float* __restrict__ sums,
                               const int*   __restrict__ counts,
                               float* __restrict__ out)
{
    if (threadIdx.x == 0 && blockIdx.x == 0) {
        float s = 0.0f;
        #pragma unroll
        for (int b = 0; b < B_COUNT; ++b) {
            const float np = fmaxf((float)counts[b], 1.0f);
            s += sums[b] / np;
        }
        out[0] = s * (1.0f / (float)B_COUNT);
    }
}

extern "C" void kernel_launch(void* const* d_in, const int* in_sizes, int n_in,
                              void* d_out, int out_size, void* d_ws, size_t ws_size,
                              hipStream_t stream) {
    const float* cls     = (const float*)d_in[0];  // (8, 65536, 80)
    const float* anchors = (const float*)d_in[1];  // (1, 65536, 2)
    const float* ann     = (const float*)d_in[2];  // (8, 64, 3)

    float* sums   = (float*)d_ws;                                  // 8 floats
    int*   counts = (int*)((char*)d_ws + B_COUNT * sizeof(float)); // 8 ints

    focal_init_ws<<<1, 32, 0, stream>>>(sums, counts);

    dim3 grid(A_COUNT / APB, B_COUNT);   // (256, 8) blocks of 256 threads
    focal_main<<<grid, 256, 0, stream>>>(cls, anchors, ann, sums, counts);

    focal_finalize<<<1, 32, 0, stream>>>(sums, counts, (float*)d_out);
}